// TrueQuantizationLayer_35708358099465
// MI455X (gfx1250) — compile-verified
//
#include <hip/hip_runtime.h>

// ---------------------------------------------------------------------------
// TrueQuantizationLayer bit-serial SAR ADC for MI455X (gfx1250).
// Memory-bound streaming kernel: ~503 MB of traffic -> ~22 us floor @ 23.3 TB/s.
// Coefficients (36 per ADC column) live in VGPRs; the small Wn table is
// broadcast into LDS per-block via the CDNA5 Tensor Data Mover (6-arg
// clang-23 builtin form, as diagnosed from the round-1 arity error).
// ---------------------------------------------------------------------------

#define NUM_ADCS   24
#define NUM_BITS   8
#define MCOEF      36                    // NUM_BITS*(NUM_BITS+1)/2
#define WN_ELEMS   (MCOEF * NUM_ADCS)    // 864 floats
#define VR_F       0.225f                // VDD / 2^3 ; VREF == VR
#define SIGMA_SQRT 0.1f                  // sqrt(NOISE_SIGMA) = sqrt(0.01)
#define DELTA_F    1e-30f

typedef float v4f __attribute__((ext_vector_type(4)));

// ---- Stage 1: Wn = W + 0.1 * noise  (written straight into d_out tail) ----
__global__ void tq_prep(const float* __restrict__ W,
                        const float* __restrict__ noise,
                        float* __restrict__ Wn_out) {
  int i = blockIdx.x * blockDim.x + threadIdx.x;
  if (i < WN_ELEMS) {
    Wn_out[i] = W[i] + noise[i] * SIGMA_SQRT;
  }
}

// ---- Stage 2: streaming SAR recurrence ------------------------------------
__global__ __launch_bounds__(384)
void tq_main(const float* __restrict__ x,
             const float* __restrict__ Wn,     // 864 floats (in d_out tail)
             float* __restrict__ q_out,        // [L*A]
             float* __restrict__ Q_out,        // [L*A*8]
             int total) {
  __shared__ float lds_wn[WN_ELEMS];

#if defined(__gfx1250__) && __has_builtin(__builtin_amdgcn_tensor_load_to_lds) && __has_builtin(__builtin_amdgcn_s_wait_tensorcnt)
  // CDNA5 Tensor Data Mover: one DMA of the 864-float table per block.
  // Issue from wave 0 only (TDM ignores EXEC; one issue per wave).
  if (threadIdx.x < 32) {
    typedef unsigned int u32x4 __attribute__((ext_vector_type(4)));
    typedef int          i32x4 __attribute__((ext_vector_type(4)));
    typedef int          i32x8 __attribute__((ext_vector_type(8)));

    unsigned long long ga = (unsigned long long)Wn;
    unsigned int lds_off  = (unsigned int)(unsigned long long)&lds_wn[0];

    // D# group 0: count=1 | lds_addr | global_addr[56:0] | type=2
    u32x4 g0;
    g0[0] = 1u;                                            // count=1, is_restore=0
    g0[1] = lds_off;                                       // lds_addr (bytes)
    g0[2] = (unsigned int)(ga & 0xFFFFFFFFull);            // global_addr[31:0]
    g0[3] = (unsigned int)((ga >> 32) & 0x1FFFFFFull)      // global_addr[56:32]
          | (2u << 30);                                    // type = 2 ("image")

    // D# group 1: data_size=4B, 1-D tensor/tile of 864 elements
    i32x8 g1;
    g1[0] = (2 << 16);                                     // data_size = 2 (4 bytes)
    g1[1] = (int)((WN_ELEMS & 0xFFFF) << 16);              // tensor_dim0[15:0]
    g1[2] = (1 << 16);                                     // tensor_dim0[31:16]=0 | tensor_dim1=1
    g1[3] = (int)(WN_ELEMS << 16);                         // tile_dim0 = 864
    g1[4] = 0;                                             // tile_dim1/2 unused
    g1[5] = WN_ELEMS;                                      // tensor_dim0_stride lo32
    g1[6] = 0;
    g1[7] = 0;

    i32x4 gz4 = {0, 0, 0, 0};                              // groups 2/3 unused (<=2D)
    i32x8 gz8 = {0, 0, 0, 0, 0, 0, 0, 0};                  // extra group (clang-23 form)
    __builtin_amdgcn_tensor_load_to_lds(g0, g1, gz4, gz4, gz8, 0);
    __builtin_amdgcn_s_wait_tensorcnt(0);
  }
#else
  for (int i = threadIdx.x; i < WN_ELEMS; i += blockDim.x) {
    lds_wn[i] = Wn[i];
  }
#endif
  __syncthreads();

  int tid    = blockIdx.x * blockDim.x + threadIdx.x;
  int stride = blockDim.x * gridDim.x;      // blockDim=384 -> stride % 24 == 0
  int a_col  = tid % NUM_ADCS;              // invariant across grid-stride iters

  // Per-column coefficients, pre-scaled by VR (VREF==VR), held in VGPRs.
  float c[MCOEF];
#pragma unroll
  for (int m = 0; m < MCOEF; ++m) {
    c[m] = lds_wn[m * NUM_ADCS + a_col] * VR_F;
  }

  for (int idx = tid; idx < total; idx += stride) {
    float xv = __builtin_nontemporal_load(x + idx);

    float s[NUM_BITS];    // sign values in {-1, 0, +1}
    float bv[NUM_BITS];   // bit values (s+1)/2
    int m = MCOEF - 1;
#pragma unroll
    for (int j = NUM_BITS - 1; j >= 0; --j) {
      float bs = c[m]; --m;                 // Wn[m]*VREF
#pragma unroll
      for (int k = j + 1; k < NUM_BITS; ++k) {
        bs = fmaf(bv[k], c[m], bs);         // (q_k+1)*0.5 * Wn[m]*VR
        --m;
      }
      float d = xv - bs + DELTA_F;
      s[j]  = (d > 0.0f) ? 1.0f : ((d < 0.0f) ? -1.0f : 0.0f);
      bv[j] = (s[j] + 1.0f) * 0.5f;
    }

    // q = sum_b bit_b * (VR * 2^b)
    float qv = 0.0f;
#pragma unroll
    for (int b = 0; b < NUM_BITS; ++b) {
      qv = fmaf(bv[b], VR_F * (float)(1 << b), qv);
    }

    __builtin_nontemporal_store(qv, q_out + idx);

    // Q[idx][0..7] as two 16-byte NT stores (32-byte aligned).
    v4f lo4 = { s[0], s[1], s[2], s[3] };
    v4f hi4 = { s[4], s[5], s[6], s[7] };
    v4f* qp = (v4f*)(Q_out + (size_t)idx * NUM_BITS);
    __builtin_nontemporal_store(lo4, qp);
    __builtin_nontemporal_store(hi4, qp + 1);
  }
}

// ---------------------------------------------------------------------------
extern "C" void kernel_launch(void* const* d_in, const int* in_sizes, int n_in,
                              void* d_out, int out_size, void* d_ws, size_t ws_size,
                              hipStream_t stream) {
  const float* x     = (const float*)d_in[0];   // [L, 24]
  const float* W     = (const float*)d_in[1];   // [36, 24]
  const float* noise = (const float*)d_in[2];   // [36, 24]

  const int total = in_sizes[0];                // L * 24 = 12,582,912

  // Output layout: q [total] | Q [total*8] | Wn [864]
  float* q_out  = (float*)d_out;
  float* Q_out  = q_out + (size_t)total;
  float* Wn_out = q_out + (size_t)total * 9;

  // Stage 1: noisy weights -> output tail (also the table TDM reads).
  tq_prep<<<(WN_ELEMS + 255) / 256, 256, 0, stream>>>(W, noise, Wn_out);

  // Stage 2: 2048 blocks x 384 threads (12 wave32 waves/block);
  // stride = 786,432 (multiple of 24) -> exactly 16 elements per thread.
  tq_main<<<2048, 384, 0, stream>>>(x, Wn_out, q_out, Q_out, total);
}